// RegionProposalNetwork_50311246905559
// MI455X (gfx1250) — compile-verified
//
#include <hip/hip_runtime.h>
#include <hip/hip_bf16.h>

// ---------------------------------------------------------------------------
// RPN proposal filtering for MI455X (gfx1250, wave32).
//   Phase 1: per-(image,level) radix-select top-k over objectness, with
//            double-buffered async global->LDS staging (ASYNCcnt path),
//            b128 (full-cacheline) async transfers when available.
//   Phase 2: gather + decode + clip + small-box mask + per-image coord max.
//   Phase 3: per-image greedy NMS, candidates register-resident, argmax via
//            wave32 shfl_xor butterflies + LDS cross-wave hop.
// ---------------------------------------------------------------------------

#define N_IMG     16
#define A_TOTAL   261888
#define KPRE      4768
#define POST_NMS  1000
#define NMS_TH    0.7f
#define MIN_SIZE  1.0f
#define NEGF      (-1.0e10f)
#define BBOX_CLIP 4.1351666f   // log(1000/16)

__constant__ int LEVEL_N[5]   = {196608, 49152, 12288, 3072, 768};
__constant__ int LEVEL_OFF[5] = {0, 196608, 245760, 258048, 261120};
__constant__ int LEVEL_K[5]   = {1000, 1000, 1000, 1000, 768};
__constant__ int SLOT_OFF[5]  = {0, 1000, 2000, 3000, 4000};

#define NSEL (N_IMG * KPRE)   // 76288

// ---- CDNA5 async global->LDS path: b128 preferred, b32 fallback, then plain.
#if defined(__gfx1250__) && \
    __has_builtin(__builtin_amdgcn_global_load_async_to_lds_b128) && \
    __has_builtin(__builtin_amdgcn_s_wait_asynccnt)
#define ASYNC_MODE 2
#elif defined(__gfx1250__) && \
    __has_builtin(__builtin_amdgcn_global_load_async_to_lds_b32) && \
    __has_builtin(__builtin_amdgcn_s_wait_asynccnt)
#define ASYNC_MODE 1
#else
#define ASYNC_MODE 0
#endif

#if ASYNC_MODE == 2
typedef int v4i_ __attribute__((ext_vector_type(4)));
typedef __attribute__((address_space(1))) v4i_ as1_v4i;
typedef __attribute__((address_space(3))) v4i_ as3_v4i;
#elif ASYNC_MODE == 1
typedef __attribute__((address_space(1))) int as1_int;
typedef __attribute__((address_space(3))) int as3_int;
#endif

__device__ __forceinline__ void async_wait0() {
#if ASYNC_MODE
  __builtin_amdgcn_s_wait_asynccnt(0);
#endif
}

// Monotone float->uint key: larger float => larger key.
__device__ __forceinline__ unsigned fkey(float f) {
  unsigned u = __float_as_uint(f);
  return (u & 0x80000000u) ? ~u : (u | 0x80000000u);
}

// ---------------------------------------------------------------------------
// Phase 0: init per-image max accumulators.
// ---------------------------------------------------------------------------
__global__ void init_kernel(unsigned* __restrict__ maxbits) {
  if (threadIdx.x < N_IMG) maxbits[threadIdx.x] = 0u;
}

// ---------------------------------------------------------------------------
// Phase 1: one block per (level, image). 4-pass radix select (8 bits/pass)
// then compaction. Histogram passes read tiles staged into LDS by async
// global->LDS loads (128-bit, full cacheline), double-buffered so the next
// tile is in flight while the current one is binned.
// ---------------------------------------------------------------------------
#define TOPK_NTH 256
#define TILE     2048
#define EPT      (TILE / TOPK_NTH)        // 8 floats/thread/tile
#define FEPT     (TILE / 4 / TOPK_NTH)    // 2 float4/thread/tile

// Stage tile starting at element `base` of src[0..n) into LDS dst[0..TILE).
// All level sizes/offsets are multiples of 4 floats -> 16B-aligned addresses.
__device__ __forceinline__ void stage_tile(const float* __restrict__ src, int n,
                                           float* dst, int base, int tid) {
#if ASYNC_MODE == 2
  const int b4 = base >> 2, n4 = n >> 2;
#pragma unroll
  for (int j = 0; j < FEPT; ++j) {
    int i4 = b4 + tid + j * TOPK_NTH;
    if (i4 < n4)
      __builtin_amdgcn_global_load_async_to_lds_b128(
          (as1_v4i*)(void*)(src + 4 * i4),
          (as3_v4i*)(void*)(dst + 4 * (tid + j * TOPK_NTH)), 0, 0);
  }
#elif ASYNC_MODE == 1
#pragma unroll
  for (int j = 0; j < EPT; ++j) {
    int i = base + tid + j * TOPK_NTH;
    if (i < n)
      __builtin_amdgcn_global_load_async_to_lds_b32(
          (as1_int*)(void*)(src + i),
          (as3_int*)(void*)(dst + tid + j * TOPK_NTH), 0, 0);
  }
#else
#pragma unroll
  for (int j = 0; j < EPT; ++j) {
    int i = base + tid + j * TOPK_NTH;
    if (i < n) dst[tid + j * TOPK_NTH] = src[i];
  }
#endif
}

__global__ __launch_bounds__(TOPK_NTH)
void topk_radix_kernel(const float* __restrict__ obj,
                       int* __restrict__ sel_idx,
                       float* __restrict__ sel_score) {
  const int level = blockIdx.x;
  const int img   = blockIdx.y;
  const int n     = LEVEL_N[level];
  const int k     = LEVEL_K[level];
  const int gOff  = LEVEL_OFF[level];
  const int slotBase = img * KPRE + SLOT_OFF[level];
  const float* __restrict__ src = obj + (size_t)img * A_TOTAL + gOff;
  const int tid = threadIdx.x;

  __shared__ __align__(16) float buf[2][TILE];
  __shared__ unsigned hist[256];
  __shared__ unsigned sPrefix, sRemK;
  __shared__ int      sCntG, sCntE;

  if (tid == 0) { sPrefix = 0u; sRemK = (unsigned)k; }
  __syncthreads();

  const int nTiles = (n + TILE - 1) / TILE;

  for (int p = 0; p < 4; ++p) {
    const int shift = 24 - 8 * p;
    const unsigned maskHigh = (p == 0) ? 0u : (0xFFFFFFFFu << (shift + 8));
    hist[tid] = 0u;
    __syncthreads();
    const unsigned pref = sPrefix & maskHigh;

    // Prologue: stage tile 0.
    stage_tile(src, n, &buf[0][0], 0, tid);
    async_wait0();
    __syncthreads();

    for (int tile = 0; tile < nTiles; ++tile) {
      const int cur = tile & 1, nxt = cur ^ 1;
      if (tile + 1 < nTiles)
        stage_tile(src, n, &buf[nxt][0], (tile + 1) * TILE, tid);

      const int base = tile * TILE;
#pragma unroll
      for (int j = 0; j < EPT; ++j) {
        int i = base + tid + j * TOPK_NTH;
        if (i < n) {
          unsigned key = fkey(buf[cur][tid + j * TOPK_NTH]);
          if ((key & maskHigh) == pref)
            atomicAdd(&hist[(key >> shift) & 255u], 1u);
        }
      }
      async_wait0();     // next buffer resident in LDS
      __syncthreads();   // all waves done reading cur before it is reissued
    }

    if (tid == 0) {
      unsigned cum = 0;
      int d = 255;
      for (; d > 0; --d) {
        if (cum + hist[d] >= sRemK) break;
        cum += hist[d];
      }
      sPrefix |= ((unsigned)d) << shift;
      sRemK -= cum;
    }
    __syncthreads();
  }

  const unsigned T = sPrefix;     // k-th largest key
  const int kEq = (int)sRemK;     // how many ==T to take
  const int nG  = k - kEq;        // count strictly greater than T
  if (tid == 0) { sCntG = 0; sCntE = 0; }
  __syncthreads();

  // Compaction: greater-than keys fill [0,nG), equal keys fill [nG, k).
  for (int i = tid; i < n; i += TOPK_NTH) {
    if (i + 4 * TOPK_NTH < n) __builtin_prefetch(src + i + 4 * TOPK_NTH, 0, 0);
    float f = src[i];
    unsigned key = fkey(f);
    if (key > T) {
      int pos = atomicAdd(&sCntG, 1);
      sel_idx[slotBase + pos]   = gOff + i;
      sel_score[slotBase + pos] = f;
    } else if (key == T) {
      int pos = atomicAdd(&sCntE, 1);
      if (pos < kEq) {
        sel_idx[slotBase + nG + pos]   = gOff + i;
        sel_score[slotBase + nG + pos] = f;
      }
    }
  }
}

// ---------------------------------------------------------------------------
// Phase 2: decode + clip + small-box mask + per-image coordinate max.
// ---------------------------------------------------------------------------
__global__ __launch_bounds__(256)
void decode_kernel(const float4* __restrict__ deltas4,
                   const float4* __restrict__ anchors4,
                   const int* __restrict__ shapes,
                   const int* __restrict__ sel_idx,
                   const float* __restrict__ sel_score,
                   float* __restrict__ scores_nms,
                   float4* __restrict__ boxes_raw,
                   unsigned* __restrict__ maxbits) {
  int gid = blockIdx.x * 256 + threadIdx.x;
  if (gid >= N_IMG * KPRE) return;
  int img = gid / KPRE;
  int a = sel_idx[gid];
  float4 an = anchors4[a];
  float4 dl = deltas4[img * A_TOTAL + a];

  float w  = an.z - an.x, h = an.w - an.y;
  float cx = an.x + 0.5f * w, cy = an.y + 0.5f * h;
  float dw = fminf(dl.z, BBOX_CLIP), dh = fminf(dl.w, BBOX_CLIP);
  float pcx = dl.x * w + cx, pcy = dl.y * h + cy;
  float pw = __expf(dw) * w, ph = __expf(dh) * h;

  float H = (float)shapes[img * 2 + 0];
  float W = (float)shapes[img * 2 + 1];
  float x1 = fminf(fmaxf(pcx - 0.5f * pw, 0.0f), W);
  float y1 = fminf(fmaxf(pcy - 0.5f * ph, 0.0f), H);
  float x2 = fminf(fmaxf(pcx + 0.5f * pw, 0.0f), W);
  float y2 = fminf(fmaxf(pcy + 0.5f * ph, 0.0f), H);

  boxes_raw[gid] = make_float4(x1, y1, x2, y2);
  bool small = ((x2 - x1) < MIN_SIZE) || ((y2 - y1) < MIN_SIZE);
  scores_nms[gid] = small ? NEGF : sel_score[gid];

  float m = fmaxf(fmaxf(x1, y1), fmaxf(x2, y2));  // coords >= 0: uint cmp ok
  atomicMax(&maxbits[img], __float_as_uint(m));
}

// ---------------------------------------------------------------------------
// Phase 3: greedy NMS, one block per image, candidates in registers.
// ---------------------------------------------------------------------------
#define NMS_NTH 512
#define CPT     10          // ceil(4768/512)
#define NWAVES  (NMS_NTH / 32)

__global__ __launch_bounds__(NMS_NTH)
void nms_kernel(const float* __restrict__ scores_in,
                const float4* __restrict__ boxes_raw,
                const unsigned* __restrict__ maxbits,
                float* __restrict__ out) {
  const int img = blockIdx.x;
  const int t = threadIdx.x;
  const int wid = t >> 5, lane = t & 31;

  __shared__ float redS[NWAVES];
  __shared__ int   redI[NWAVES];
  __shared__ float sBestS;
  __shared__ int   sBestI;
  __shared__ float bbx1, bby1, bbx2, bby2, bbar;
  __shared__ int   keepArr[POST_NMS];

  const float offmul = __uint_as_float(maxbits[img]) + 1.0f;

  float sc[CPT], x1v[CPT], y1v[CPT], x2v[CPT], y2v[CPT], ar[CPT];
  int   sl[CPT];
#pragma unroll
  for (int c = 0; c < CPT; ++c) {
    int slot = t + c * NMS_NTH;
    if (slot < KPRE) {
      float4 b = boxes_raw[img * KPRE + slot];
      float off = (float)(slot / 1000) * offmul;   // batched-NMS level offset
      x1v[c] = b.x + off; y1v[c] = b.y + off;
      x2v[c] = b.z + off; y2v[c] = b.w + off;
      ar[c]  = (x2v[c] - x1v[c]) * (y2v[c] - y1v[c]);
      sc[c]  = scores_in[img * KPRE + slot];
      sl[c]  = slot;
    } else {
      sc[c] = -3.0e38f; sl[c] = 0x7fffffff;
      x1v[c] = y1v[c] = x2v[c] = y2v[c] = ar[c] = 0.0f;
    }
  }

  for (int it = 0; it < POST_NMS; ++it) {
    // local argmax (prefer lower slot on ties, like jnp.argmax)
    float bs = sc[0]; int bi = sl[0];
#pragma unroll
    for (int c = 1; c < CPT; ++c)
      if (sc[c] > bs || (sc[c] == bs && sl[c] < bi)) { bs = sc[c]; bi = sl[c]; }
    // wave32 butterfly
#pragma unroll
    for (int m = 16; m > 0; m >>= 1) {
      float os = __shfl_xor(bs, m);
      int   oi = __shfl_xor(bi, m);
      if (os > bs || (os == bs && oi < bi)) { bs = os; bi = oi; }
    }
    if (lane == 0) { redS[wid] = bs; redI[wid] = bi; }
    __syncthreads();                          // B1
    if (t == 0) {
      float gs = redS[0]; int gi = redI[0];
      for (int w = 1; w < NWAVES; ++w)
        if (redS[w] > gs || (redS[w] == gs && redI[w] < gi)) { gs = redS[w]; gi = redI[w]; }
      sBestS = gs; sBestI = gi;
      keepArr[it] = (gs > NEGF * 0.5f) ? gi : -1;
    }
    __syncthreads();                          // B2
    const float gs = sBestS;
    const int   gi = sBestI;
    const bool  ok = gs > NEGF * 0.5f;
    if (ok) {
#pragma unroll
      for (int c = 0; c < CPT; ++c)
        if (sl[c] == gi) { bbx1 = x1v[c]; bby1 = y1v[c]; bbx2 = x2v[c]; bby2 = y2v[c]; bbar = ar[c]; }
    }
    __syncthreads();                          // B3
    if (ok) {
      const float jx1 = bbx1, jy1 = bby1, jx2 = bbx2, jy2 = bby2, jar = bbar;
#pragma unroll
      for (int c = 0; c < CPT; ++c) {
        float ix1 = fmaxf(x1v[c], jx1), iy1 = fmaxf(y1v[c], jy1);
        float ix2 = fminf(x2v[c], jx2), iy2 = fminf(y2v[c], jy2);
        float inter = fmaxf(ix2 - ix1, 0.0f) * fmaxf(iy2 - iy1, 0.0f);
        float iou = inter / (ar[c] + jar - inter + 1e-9f);
        if (iou > NMS_TH || sl[c] == gi) sc[c] = NEGF;
      }
    }
  }
  __syncthreads();

  // Emit padded output: invalid slots zeroed.
  for (int i = t; i < POST_NMS; i += NMS_NTH) {
    int kp = keepArr[i];
    float4 o = make_float4(0.0f, 0.0f, 0.0f, 0.0f);
    if (kp >= 0) o = boxes_raw[img * KPRE + kp];
    reinterpret_cast<float4*>(out)[img * POST_NMS + i] = o;
  }
}

// ---------------------------------------------------------------------------
extern "C" void kernel_launch(void* const* d_in, const int* in_sizes, int n_in,
                              void* d_out, int out_size, void* d_ws, size_t ws_size,
                              hipStream_t stream) {
  const float* obj     = (const float*)d_in[0];  // [16, 261888]
  const float* deltas  = (const float*)d_in[1];  // [16, 261888, 4]
  const float* anchors = (const float*)d_in[2];  // [261888, 4]
  const int*   shapes  = (const int*)d_in[3];    // [16, 2]
  float* out = (float*)d_out;                    // [16, 1000, 4]

  char* ws = (char*)d_ws;
  int*      sel_idx    = (int*)ws;                                   // NSEL ints
  float*    sel_score  = (float*)(ws + (size_t)NSEL * 4);            // NSEL f32
  float*    scores_nms = (float*)(ws + (size_t)NSEL * 8);            // NSEL f32
  float4*   boxes_raw  = (float4*)(ws + (size_t)NSEL * 12);          // NSEL f4 (16B aligned)
  unsigned* maxbits    = (unsigned*)(ws + (size_t)NSEL * 12 + (size_t)NSEL * 16);

  init_kernel<<<1, 32, 0, stream>>>(maxbits);
  topk_radix_kernel<<<dim3(5, N_IMG), TOPK_NTH, 0, stream>>>(obj, sel_idx, sel_score);
  decode_kernel<<<(N_IMG * KPRE + 255) / 256, 256, 0, stream>>>(
      (const float4*)deltas, (const float4*)anchors, shapes,
      sel_idx, sel_score, scores_nms, boxes_raw, maxbits);
  nms_kernel<<<N_IMG, NMS_NTH, 0, stream>>>(scores_nms, boxes_raw, maxbits, out);
}